// NNConvNet_12927851561053
// MI455X (gfx1250) — compile-verified
//
#include <hip/hip_runtime.h>

typedef __attribute__((ext_vector_type(2))) float v2f;
typedef __attribute__((ext_vector_type(8))) float v8f;

#define N_NODES  10000
#define N_EDGES  30000
#define N_GRAPHS 32

// ---------------------------------------------------------------- utilities
__global__ void zero_kernel(float* __restrict__ p, int n) {
  int i = blockIdx.x * blockDim.x + threadIdx.x;
  if (i < n) p[i] = 0.0f;
}

__global__ void deg_kernel(const long long* __restrict__ ei, float* __restrict__ deg) {
  int e = blockIdx.x * blockDim.x + threadIdx.x;
  if (e < N_EDGES) atomicAdd(&deg[(int)ei[N_EDGES + e]], 1.0f);
}

// Repack We[8, din*dout], be[din*dout], root[din, dout] into one GEMM B-matrix
// W'[din x nco], nco = 10*dout:
//   cols [0, 8*dout)      : W'[i, k*dout+o] = We[k, i*dout+o]   (ea-weighted part)
//   cols [8*dout, 9*dout) : W'[i, ..+o]     = be[i*dout+o]      (edge bias term)
//   cols [9*dout, 10*dout): W'[i, ..+o]     = root[i*dout+o]    (root transform)
__global__ void repack_kernel(const float* __restrict__ We, const float* __restrict__ be,
                              const float* __restrict__ root, float* __restrict__ Wp,
                              int din, int dout, int nco) {
  int t = blockIdx.x * blockDim.x + threadIdx.x;
  if (t >= din * nco) return;
  int i = t / nco, c = t % nco;
  float v;
  if (c < 8 * dout)            v = We[(c / dout) * (din * dout) + i * dout + (c % dout)];
  else if (c < 9 * dout)       v = be[i * dout + (c - 8 * dout)];
  else                         v = root[i * dout + (c - 9 * dout)];
  Wp[(size_t)i * nco + c] = v;
}

// ------------------------------------------------ dense fp32 WMMA GEMM core
// Z[M x nco] = A[M x K] @ B[K x nco].
// Register-blocked: each wave32 computes a 16x80 strip (one M-tile x five
// N-tiles, five v8f accumulators) via V_WMMA_F32_16X16X4_F32. K is a template
// parameter so the k-loop fully unrolls into a dense WMMA chain; the A
// fragment is loaded once per k-step (8B vector load) and reused across the
// five tiles.
template <int K>
__global__ void gemm_wmma_kernel(const float* __restrict__ A, const float* __restrict__ B,
                                 float* __restrict__ Z, int Mtiles, int NT5, int nco) {
  int wave = blockIdx.x * (blockDim.x >> 5) + (threadIdx.x >> 5);
  if (wave >= Mtiles * NT5) return;               // wave-uniform: EXEC stays all-ones
  int tm  = wave / NT5;
  int tn0 = (wave % NT5) * 5;
  int lane = threadIdx.x & 31;
  int r  = lane & 15;            // row within A-tile / col within B-tile
  int kh = (lane >> 4) << 1;     // half-wave K offset: 0 or 2

  const float* arow  = A + (size_t)(tm * 16 + r) * K;      // 8B aligned at +kh
  const float* bbase = B + (size_t)tn0 * 16 + r;
  v8f acc[5] = {};
#pragma unroll
  for (int k0 = 0; k0 < K; k0 += 4) {
    // A 16x4 fragment: lanes 0-15 hold K=k0+{0,1}, lanes 16-31 K=k0+{2,3}
    v2f a = *(const v2f*)(arow + k0 + kh);
    const float* brow0 = bbase + (size_t)(k0 + kh) * nco;
    const float* brow1 = brow0 + nco;
#pragma unroll
    for (int j = 0; j < 5; ++j) {
      // B 4x16 fragment: same half-wave K striping, one row per reg
      v2f b;
      b.x = brow0[j * 16];
      b.y = brow1[j * 16];
      acc[j] = __builtin_amdgcn_wmma_f32_16x16x4_f32(
          /*neg_a=*/false, a, /*neg_b=*/false, b,
          /*c_mod=*/(short)0, acc[j], /*reuse_a=*/false, /*reuse_b=*/false);
    }
  }
  // C/D layout: reg v -> row v (lanes 0-15) or v+8 (lanes 16-31), col = lane&15
  int rowbase = tm * 16 + ((lane >> 4) << 3);
#pragma unroll
  for (int j = 0; j < 5; ++j) {
    int col = (tn0 + j) * 16 + r;
#pragma unroll
    for (int v = 0; v < 8; ++v)
      Z[(size_t)(rowbase + v) * nco + col] = acc[j][v];
  }
}

// ------------------------------------------------------- edge scatter phase
// msg[e,o] = Z[src, 8*dout+o] + sum_k ea[e,k]*Z[src, k*dout+o];  agg[dst,o] += msg
__global__ void edge_kernel(const float* __restrict__ ea, const long long* __restrict__ ei,
                            const float* __restrict__ Z, float* __restrict__ agg,
                            int dout, int nco) {
  int t = blockIdx.x * blockDim.x + threadIdx.x;
  if (t >= N_EDGES * dout) return;
  int e = t / dout, o = t % dout;
  int src = (int)ei[e];
  int dst = (int)ei[N_EDGES + e];
  const float* z = Z + (size_t)src * nco;
  const float* a = ea + (size_t)e * 8;
  float m = z[8 * dout + o];           // be-term
#pragma unroll
  for (int k = 0; k < 8; ++k) m += a[k] * z[k * dout + o];
  atomicAdd(&agg[(size_t)dst * dout + o], m);
}

// h_next = relu(agg/deg + (h@root) + b)   (root part lives in Z cols [9*dout, 10*dout))
__global__ void node_kernel(const float* __restrict__ agg, const float* __restrict__ Z,
                            const float* __restrict__ deg, const float* __restrict__ b,
                            float* __restrict__ h, int dout, int nco) {
  int t = blockIdx.x * blockDim.x + threadIdx.x;
  if (t >= N_NODES * dout) return;
  int n = t / dout, o = t % dout;
  float d = deg[n]; if (d < 1.0f) d = 1.0f;
  float v = agg[t] / d + Z[(size_t)n * nco + 9 * dout + o] + b[o];
  h[t] = v > 0.0f ? v : 0.0f;
}

// segment_max over nonneg (post-ReLU) values: uint atomicMax on the float bits
__global__ void pool_kernel(const float* __restrict__ h, const long long* __restrict__ batch,
                            float* __restrict__ pool) {
  int t = blockIdx.x * blockDim.x + threadIdx.x;
  if (t >= N_NODES * 128) return;
  int n = t >> 7, o = t & 127;
  int g = (int)batch[n];
  atomicMax((unsigned int*)&pool[g * 128 + o], __float_as_uint(h[t]));
}

// tiny head: relu(pool@fc1 + b1) @ fc2 + b2   (32x128 -> 32x64 -> 32x10)
__global__ void mlp_kernel(const float* __restrict__ pool,
                           const float* __restrict__ w1, const float* __restrict__ b1,
                           const float* __restrict__ w2, const float* __restrict__ b2,
                           float* __restrict__ out) {
  __shared__ float hid[N_GRAPHS * 64];
  for (int idx = threadIdx.x; idx < N_GRAPHS * 64; idx += blockDim.x) {
    int g = idx >> 6, j = idx & 63;
    float s = b1[j];
    for (int i = 0; i < 128; ++i) s += pool[g * 128 + i] * w1[i * 64 + j];
    hid[idx] = s > 0.0f ? s : 0.0f;
  }
  __syncthreads();
  for (int idx = threadIdx.x; idx < N_GRAPHS * 10; idx += blockDim.x) {
    int g = idx / 10, j = idx % 10;
    float s = b2[j];
    for (int i = 0; i < 64; ++i) s += hid[g * 64 + i] * w2[i * 10 + j];
    out[idx] = s;
  }
}

// --------------------------------------------------------------- dispatcher
static inline int cdiv(int a, int b) { return (a + b - 1) / b; }

extern "C" void kernel_launch(void* const* d_in, const int* in_sizes, int n_in,
                              void* d_out, int out_size, void* d_ws, size_t ws_size,
                              hipStream_t stream) {
  (void)in_sizes; (void)n_in; (void)out_size; (void)ws_size;

  const float*     x     = (const float*)d_in[0];
  const float*     ea    = (const float*)d_in[1];
  const long long* ei    = (const long long*)d_in[2];
  const long long* batch = (const long long*)d_in[3];
  const float *We[4], *be[4], *root[4], *bb[4];
  for (int l = 0; l < 4; ++l) {
    We[l]   = (const float*)d_in[4 + 4 * l];
    be[l]   = (const float*)d_in[5 + 4 * l];
    root[l] = (const float*)d_in[6 + 4 * l];
    bb[l]   = (const float*)d_in[7 + 4 * l];
  }
  const float* fc1w = (const float*)d_in[20];
  const float* fc1b = (const float*)d_in[21];
  const float* fc2w = (const float*)d_in[22];
  const float* fc2b = (const float*)d_in[23];

  // workspace carve-up (floats)
  float* ws  = (float*)d_ws;
  size_t off = 0;
  float* deg  = ws + off; off += N_NODES;
  float* hA   = ws + off; off += (size_t)N_NODES * 128;
  float* hB   = ws + off; off += (size_t)N_NODES * 128;
  float* Z    = ws + off; off += (size_t)N_NODES * 1280;   // max nco = 10*128
  float* Wp   = ws + off; off += (size_t)64 * 1280;        // max din x nco
  float* agg  = ws + off; off += (size_t)N_NODES * 128;
  float* pool = ws + off; off += (size_t)N_GRAPHS * 128;

  zero_kernel<<<cdiv(N_NODES, 256), 256, 0, stream>>>(deg, N_NODES);
  deg_kernel<<<cdiv(N_EDGES, 256), 256, 0, stream>>>(ei, deg);

  const int dins[4]  = {16, 32, 48, 64};
  const int douts[4] = {32, 48, 64, 128};
  const float* hin = x;
  float* bufs[2] = {hA, hB};
  for (int l = 0; l < 4; ++l) {
    int din = dins[l], dout = douts[l], nco = 10 * dout;
    repack_kernel<<<cdiv(din * nco, 256), 256, 0, stream>>>(
        We[l], be[l], root[l], Wp, din, dout, nco);

    int Mtiles = N_NODES / 16;        // 10000 = 625 * 16 exactly
    int Ntiles = nco / 16;            // {320,480,640,1280}/16 -> {20,30,40,80}
    int NT5    = Ntiles / 5;          // all divisible by 5
    int waves  = Mtiles * NT5;
    int blocks = cdiv(waves, 8);      // 256 threads = 8 waves per block
    switch (din) {
      case 16: gemm_wmma_kernel<16><<<blocks, 256, 0, stream>>>(hin, Wp, Z, Mtiles, NT5, nco); break;
      case 32: gemm_wmma_kernel<32><<<blocks, 256, 0, stream>>>(hin, Wp, Z, Mtiles, NT5, nco); break;
      case 48: gemm_wmma_kernel<48><<<blocks, 256, 0, stream>>>(hin, Wp, Z, Mtiles, NT5, nco); break;
      default: gemm_wmma_kernel<64><<<blocks, 256, 0, stream>>>(hin, Wp, Z, Mtiles, NT5, nco); break;
    }

    zero_kernel<<<cdiv(N_NODES * dout, 256), 256, 0, stream>>>(agg, N_NODES * dout);
    edge_kernel<<<cdiv(N_EDGES * dout, 256), 256, 0, stream>>>(ea, ei, Z, agg, dout, nco);
    float* hout = bufs[l & 1];
    node_kernel<<<cdiv(N_NODES * dout, 256), 256, 0, stream>>>(
        agg, Z, deg, bb[l], hout, dout, nco);
    hin = hout;
  }

  zero_kernel<<<cdiv(N_GRAPHS * 128, 256), 256, 0, stream>>>(pool, N_GRAPHS * 128);
  pool_kernel<<<cdiv(N_NODES * 128, 256), 256, 0, stream>>>(hin, batch, pool);
  mlp_kernel<<<1, 256, 0, stream>>>(pool, fc1w, fc1b, fc2w, fc2b, (float*)d_out);
}